// CTRNN_32212254720570
// MI455X (gfx1250) — compile-verified
//
#include <hip/hip_runtime.h>

// CTRNN on MI455X (gfx1250, wave32).
//  Phase 1: x_proj = input @ W_in^T + b_in + b_hh  -> written in place into d_out.
//  Phase 2: 1024 sequential step kernels: h = 0.8*h + 0.2*relu(x_proj + h @ W_hh^T).
// Full-f32 matrix math via V_WMMA_F32_16X16X4_F32 (recurrence over 1024 steps
// would accumulate bf16 rounding error; phase 2 is latency-bound so the f32
// WMMA path costs nothing at the wall clock).

typedef float v2f __attribute__((ext_vector_type(2)));
typedef float v8f __attribute__((ext_vector_type(8)));

#define CTRNN_T 1024
#define CTRNN_B 128
#define CTRNN_I 128
#define CTRNN_H 512
#define CTRNN_ALPHA 0.2f

// Fragment loader for V_WMMA_F32_16X16X4_F32.
// A-matrix (16x4, MxK) ISA layout: lanes 0-15 hold {K=k, K=k+1} of row (lane&15),
// lanes 16-31 hold {K=k+2, K=k+3}. The B-matrix (4x16, KxN) of our GEMMs is
// W^T with W stored row-major [N,K], which maps to the exact same per-lane
// addressing with "row" = N. k is a multiple of 4 and strides are even, so the
// float2 load is 8-byte aligned.
__device__ __forceinline__ v2f frag_ld(const float* __restrict__ p,
                                       int row, int ldk, int k, int lane) {
  return *reinterpret_cast<const v2f*>(p + (size_t)row * ldk +
                                       (k + ((lane >> 4) << 1)));
}

__device__ __forceinline__ v8f wmma_f32(v2f a, v2f b, v8f c) {
  // (neg_a, A, neg_b, B, c_mod, C, reuse_a, reuse_b)
  return __builtin_amdgcn_wmma_f32_16x16x4_f32(false, a, false, b, (short)0, c,
                                               false, false);
}

// ---------------------------------------------------------------------------
// Phase 1: Y[m, n] = sum_k X[m, k] * Win[n, k] + bin[n] + bhh[n]
//   M = T*B = 131072, K = I = 128, N = H = 512.
// One wave computes a 16x64 strip: 4 accumulators share each A fragment.
// ---------------------------------------------------------------------------
__global__ __launch_bounds__(128) void ctrnn_xproj_kernel(
    const float* __restrict__ X, const float* __restrict__ Win,
    const float* __restrict__ bin, const float* __restrict__ bhh,
    float* __restrict__ Y) {
  const int lane = threadIdx.x & 31;
  const int gw = blockIdx.x * 4 + (threadIdx.x >> 5);
  const int m0 = (gw >> 3) << 4;  // 8192 M-tiles of 16
  const int n0 = (gw & 7) << 6;   // 8 N-strips of 64

  v8f c0 = {}, c1 = {}, c2 = {}, c3 = {};
  const int ra = m0 + (lane & 15);
  const int rb = n0 + (lane & 15);
#pragma unroll 4
  for (int k = 0; k < CTRNN_I; k += 4) {
    v2f a = frag_ld(X, ra, CTRNN_I, k, lane);
    v2f b0 = frag_ld(Win, rb + 0, CTRNN_I, k, lane);
    v2f b1 = frag_ld(Win, rb + 16, CTRNN_I, k, lane);
    v2f b2 = frag_ld(Win, rb + 32, CTRNN_I, k, lane);
    v2f b3 = frag_ld(Win, rb + 48, CTRNN_I, k, lane);
    c0 = wmma_f32(a, b0, c0);
    c1 = wmma_f32(a, b1, c1);
    c2 = wmma_f32(a, b2, c2);
    c3 = wmma_f32(a, b3, c3);
  }
  // C/D layout: VGPR j = row m0+j (lanes 0-15) / m0+8+j (lanes 16-31), col = lane&15.
  const int col = lane & 15;
  const int rbase = m0 + ((lane >> 4) << 3);
  const float bias0 = bin[n0 + col] + bhh[n0 + col];
  const float bias1 = bin[n0 + 16 + col] + bhh[n0 + 16 + col];
  const float bias2 = bin[n0 + 32 + col] + bhh[n0 + 32 + col];
  const float bias3 = bin[n0 + 48 + col] + bhh[n0 + 48 + col];
#pragma unroll
  for (int j = 0; j < 8; ++j) {
    const size_t r = (size_t)(rbase + j) * CTRNN_H;
    Y[r + n0 + col] = c0[j] + bias0;
    Y[r + n0 + 16 + col] = c1[j] + bias1;
    Y[r + n0 + 32 + col] = c2[j] + bias2;
    Y[r + n0 + 48 + col] = c3[j] + bias3;
  }
}

// ---------------------------------------------------------------------------
// Phase 2 (one timestep): Yt holds x_proj[t] on entry, h[t] on exit.
//   pre  = Yt[b,h] + sum_j Hp[b,j] * Whh[h,j]
//   Yt   = 0.8*Hp + 0.2*relu(pre)
// M = B = 128, N = K = H = 512. 64 waves (16 blocks) cover the 8x8 strip grid.
// W_hh (1 MB) + h (256 KB) stay hot in the 192 MB L2 across all 1024 launches.
// ---------------------------------------------------------------------------
__global__ __launch_bounds__(128) void ctrnn_step_kernel(
    const float* __restrict__ Hp, const float* __restrict__ Whh,
    float* __restrict__ Yt) {
  const int lane = threadIdx.x & 31;
  const int gw = blockIdx.x * 4 + (threadIdx.x >> 5);
  const int m0 = (gw >> 3) << 4;  // 8 M-tiles of 16
  const int n0 = (gw & 7) << 6;   // 8 N-strips of 64

  v8f c0 = {}, c1 = {}, c2 = {}, c3 = {};
  const int ra = m0 + (lane & 15);
  const int rb = n0 + (lane & 15);
#pragma unroll 4
  for (int k = 0; k < CTRNN_H; k += 4) {
    v2f a = frag_ld(Hp, ra, CTRNN_H, k, lane);
    v2f b0 = frag_ld(Whh, rb + 0, CTRNN_H, k, lane);
    v2f b1 = frag_ld(Whh, rb + 16, CTRNN_H, k, lane);
    v2f b2 = frag_ld(Whh, rb + 32, CTRNN_H, k, lane);
    v2f b3 = frag_ld(Whh, rb + 48, CTRNN_H, k, lane);
    c0 = wmma_f32(a, b0, c0);
    c1 = wmma_f32(a, b1, c1);
    c2 = wmma_f32(a, b2, c2);
    c3 = wmma_f32(a, b3, c3);
  }
  const int col = lane & 15;
  const int rbase = m0 + ((lane >> 4) << 3);
#pragma unroll
  for (int j = 0; j < 8; ++j) {
    const size_t r = (size_t)(rbase + j) * CTRNN_H;
#define CTRNN_EPI(ACC, OFF)                                                   \
    {                                                                         \
      const size_t idx = r + n0 + (OFF) + col;                                \
      const float pre = ACC[j] + Yt[idx];                                     \
      Yt[idx] = Hp[idx] * (1.0f - CTRNN_ALPHA) +                              \
                fmaxf(pre, 0.0f) * CTRNN_ALPHA;                               \
    }
    CTRNN_EPI(c0, 0)
    CTRNN_EPI(c1, 16)
    CTRNN_EPI(c2, 32)
    CTRNN_EPI(c3, 48)
#undef CTRNN_EPI
  }
}

extern "C" void kernel_launch(void* const* d_in, const int* in_sizes, int n_in,
                              void* d_out, int out_size, void* d_ws,
                              size_t ws_size, hipStream_t stream) {
  (void)in_sizes; (void)n_in; (void)out_size; (void)d_ws; (void)ws_size;
  const float* X = (const float*)d_in[0];    // [T,B,I]
  const float* Win = (const float*)d_in[1];  // [H,I]
  const float* bin = (const float*)d_in[2];  // [H]
  const float* Whh = (const float*)d_in[3];  // [H,H]
  const float* bhh = (const float*)d_in[4];  // [H]
  const float* h0 = (const float*)d_in[5];   // [B,H]
  float* out = (float*)d_out;                // [T,B,H] output ++ [B,H] hidden

  const size_t stepElems = (size_t)CTRNN_B * CTRNN_H;

  // Phase 1: all input projections in one big WMMA GEMM, in place into d_out.
  {
    const int waves = (CTRNN_T * CTRNN_B / 16) * (CTRNN_H / 64);  // 65536
    ctrnn_xproj_kernel<<<waves / 4, 128, 0, stream>>>(X, Win, bin, bhh, out);
  }
  // Phase 2: sequential scan, one kernel per timestep (graph-friendly).
  for (int t = 0; t < CTRNN_T; ++t) {
    const float* hprev = (t == 0) ? h0 : out + (size_t)(t - 1) * stepElems;
    ctrnn_step_kernel<<<16, 128, 0, stream>>>(hprev, Whh,
                                              out + (size_t)t * stepElems);
  }
  // Final hidden state = output at t = T-1.
  hipMemcpyAsync(out + (size_t)CTRNN_T * stepElems,
                 out + (size_t)(CTRNN_T - 1) * stepElems,
                 stepElems * sizeof(float), hipMemcpyDeviceToDevice, stream);
}